// Resizing1D_6296422056190
// MI455X (gfx1250) — compile-verified
//
#include <hip/hip_runtime.h>

typedef float v2f __attribute__((ext_vector_type(2)));
typedef float v8f __attribute__((ext_vector_type(8)));
typedef int   v4i __attribute__((ext_vector_type(4)));

#define TARGET   1000
#define T_TILES  63          // ceil(1000/16)
#define KMAX     64          // max input rows spanned by a 16-pos tile (scale<=4.096)
#define LDSS     132         // padded LDS row stride (floats) to break bank conflicts

#define AS_GLOBAL __attribute__((address_space(1)))
#define AS_LDS    __attribute__((address_space(3)))

#if defined(__has_builtin)
#if __has_builtin(__builtin_amdgcn_global_load_async_to_lds_b128)
#define HAVE_ASYNC_LDS 1
#endif
#endif

// ---- kernel 1: lengths[b] = popcount(mask row) (mask is a prefix mask) ----
__global__ void lengths_kernel(const int* __restrict__ mask,
                               int* __restrict__ lengths, int W) {
    int b = blockIdx.x;
    const int* row = mask + (long)b * W;
    int sum = 0;
    for (int i = threadIdx.x; i < W; i += 32) sum += (row[i] != 0) ? 1 : 0;
    for (int off = 16; off > 0; off >>= 1) sum += __shfl_down(sum, off, 32);
    if (threadIdx.x == 0) lengths[b] = sum;
}

// ---- kernel 2: per-tile sparse-interp matmul via V_WMMA_F32_16X16X4_F32 ----
__device__ __forceinline__ void taps(int t, float scale, int L,
                                     int& i0, int& i1, float& w) {
    float pos = ((float)t + 0.5f) * scale - 0.5f;
    pos = fminf(fmaxf(pos, 0.0f), (float)(L - 1));
    i0 = (int)floorf(pos);
    i1 = min(i0 + 1, L - 1);
    w  = pos - (float)i0;
}

__global__ __launch_bounds__(256) void resize_wmma_kernel(
    const float* __restrict__ x, const int* __restrict__ lengths,
    float* __restrict__ out, int W, int C) {
    __shared__ float sX[KMAX * LDSS];

    const int b  = blockIdx.x / T_TILES;
    const int tt = blockIdx.x % T_TILES;
    const int t0 = tt * 16;
    const int L  = lengths[b];
    const float scale = (float)L / (float)TARGET;

    // block-uniform row range for this tile (pos is monotonic in t)
    int i0a, i1a, i0b, i1b; float wa, wb;
    taps(t0, scale, L, i0a, i1a, wa);
    taps(min(t0 + 15, TARGET - 1), scale, L, i0b, i1b, wb);
    const int rmin = i0a;
    int kNum = i1b - rmin + 1;
    if (kNum > KMAX) kNum = KMAX;
    const int kPad = (kNum + 3) & ~3;

    // ---- stage rows [rmin, rmin+kNum) into LDS; zero-fill pad rows ----
    const int tid = threadIdx.x;
    const float* xb = x + ((long)b * W + rmin) * C;
    for (int idx = tid; idx < kNum * 32; idx += 256) {
        int r = idx >> 5, c4 = (idx & 31) << 2;
        const float* gp = xb + (long)r * C + c4;
        float* lp = &sX[r * LDSS + c4];
#ifdef HAVE_ASYNC_LDS
        // direct global -> LDS, no VGPR bounce; tracked by ASYNCcnt
        __builtin_amdgcn_global_load_async_to_lds_b128(
            (AS_GLOBAL v4i*)gp, (AS_LDS v4i*)lp, 0, 0);
#else
        float4 v = *(const float4*)gp;
        lp[0] = v.x; lp[1] = v.y; lp[2] = v.z; lp[3] = v.w;
#endif
    }
    for (int idx = kNum * 32 + tid; idx < kPad * 32; idx += 256) {
        int r = idx >> 5, c4 = (idx & 31) << 2;
        float* lp = &sX[r * LDSS + c4];
        lp[0] = 0.f; lp[1] = 0.f; lp[2] = 0.f; lp[3] = 0.f;
    }
#ifdef HAVE_ASYNC_LDS
#if __has_builtin(__builtin_amdgcn_s_wait_asynccnt)
    __builtin_amdgcn_s_wait_asynccnt(0);
#else
    asm volatile("s_wait_asynccnt 0x0" ::: "memory");
#endif
#endif
    __syncthreads();

    const int wave = tid >> 5;          // 8 waves -> 8 channel slabs of 16
    const int lane = tid & 31;
    const int half = lane >> 4;         // K-pair selector for A and B fragments
    const int n    = lane & 15;         // A: M (output pos); B/D: N (channel)
    const int c0   = wave * 16;

    // this lane's bilinear taps (A-matrix row M = n)
    int i0, i1; float w;
    taps(min(t0 + n, TARGET - 1), scale, L, i0, i1, w);
    i0 -= rmin; i1 -= rmin;
    const float w0 = 1.0f - w, w1 = w;

    // software-pipelined K loop: LDS load of chunk k+4 issued before WMMA of k
    const int bBase = c0 + n;
    int kA = half * 2;
    v2f Bf = { sX[kA * LDSS + bBase], sX[(kA + 1) * LDSS + bBase] };
    v8f acc = {};
    for (int kk = 0; kk < kPad; kk += 4) {
        int knext = kk + 4;
        if (knext > kPad - 4) knext = kPad - 4;   // clamp: dummy reload, discarded
        const int kB = knext + half * 2;
        v2f Bn = { sX[kB * LDSS + bBase], sX[(kB + 1) * LDSS + bBase] };

        const int k0 = kk + half * 2;
        float a0 = ((k0     == i0) ? w0 : 0.0f) + ((k0     == i1) ? w1 : 0.0f);
        float a1 = ((k0 + 1 == i0) ? w0 : 0.0f) + ((k0 + 1 == i1) ? w1 : 0.0f);
        v2f A = {a0, a1};
        acc = __builtin_amdgcn_wmma_f32_16x16x4_f32(
            /*neg_a=*/false, A, /*neg_b=*/false, Bf,
            /*c_mod=*/(short)0, acc, /*reuse_a=*/false, /*reuse_b=*/false);
        Bf = Bn;
    }

    // ---- store D: lane N = n, VGPR v -> M = 8*half + v ----
    // block-uniform full/partial split: only the last t-tile is partial
    float* p = out + ((long)b * TARGET + (t0 + half * 8)) * C + c0 + n;
    if (t0 + 16 <= TARGET) {
#pragma unroll
        for (int v = 0; v < 8; ++v) p[(long)v * C] = acc[v];
    } else {
#pragma unroll
        for (int v = 0; v < 8; ++v) {
            if (t0 + half * 8 + v < TARGET) p[(long)v * C] = acc[v];
        }
    }
}

extern "C" void kernel_launch(void* const* d_in, const int* in_sizes, int n_in,
                              void* d_out, int out_size, void* d_ws, size_t ws_size,
                              hipStream_t stream) {
    const float* x    = (const float*)d_in[0];
    const int*   mask = (const int*)d_in[1];
    float*       out  = (float*)d_out;

    // derive shapes: out = [B, 1000, C], mask = [B, W], x = [B, W, C]
    const int maskN = in_sizes[1];
    const int xN    = in_sizes[0];
    const int C = xN / maskN;                 // 128
    const int B = out_size / (TARGET * C);    // 64
    const int W = maskN / B;                  // 4096

    int* lengths = (int*)d_ws;                // B ints of scratch

    lengths_kernel<<<B, 32, 0, stream>>>(mask, lengths, W);
    resize_wmma_kernel<<<B * T_TILES, 256, 0, stream>>>(x, lengths, out, W, C);
}